// GraphConvolution_5806795784424
// MI455X (gfx1250) — compile-verified
//
#include <hip/hip_runtime.h>

#define N_NODES 50000
#define N_EDGES 800000
#define DIM 128
#define NEG_SLOPE 0.2f
#define L2EPS 1e-12f

typedef __attribute__((ext_vector_type(2))) float v2f;
typedef __attribute__((ext_vector_type(8))) float v8f;

// ---------------------------------------------------------------------------
// Kernel 1: zero the scatter accumulator (acc[N*D] ++ cnt[N], contiguous)
// ---------------------------------------------------------------------------
__global__ void zero_ws_kernel(float* __restrict__ buf, int total) {
    int i = blockIdx.x * blockDim.x + threadIdx.x;
    if (i < total) buf[i] = 0.0f;
}

// ---------------------------------------------------------------------------
// Kernel 2: wave-per-directed-edge scatter-add (gather float4, 4 f32 atomics)
// edges is [2, N_EDGES] int64: row0 = src, row1 = dst. Undirected => both dirs.
// ---------------------------------------------------------------------------
__global__ void scatter_accum_kernel(const float* __restrict__ node_fts,
                                     const long long* __restrict__ edges,
                                     float* __restrict__ acc,
                                     float* __restrict__ cnt) {
    const unsigned tid  = blockIdx.x * blockDim.x + threadIdx.x;
    const unsigned e    = tid >> 5;          // one wave32 per directed edge
    const unsigned lane = tid & 31u;
    if (e >= 2u * N_EDGES) return;

    long long s, d;
    if (e < N_EDGES) {                       // forward direction
        s = edges[e];
        d = edges[N_EDGES + e];
    } else {                                 // flipped direction
        const unsigned e2 = e - N_EDGES;
        s = edges[N_EDGES + e2];
        d = edges[e2];
    }

    const float4 v = *(const float4*)(node_fts + (size_t)d * DIM + lane * 4u);
    float* dst = acc + (size_t)s * DIM + lane * 4u;
    atomicAdd(dst + 0, v.x);
    atomicAdd(dst + 1, v.y);
    atomicAdd(dst + 2, v.z);
    atomicAdd(dst + 3, v.w);
    if (lane == 0) atomicAdd(cnt + s, 1.0f);
}

// ---------------------------------------------------------------------------
// Kernel 3: fused  self-GEMM + mean + neigh-GEMM + LeakyReLU + L2-normalize
// Block = 256 threads = 8 waves; block owns 16 node rows; wave w owns output
// columns [16w, 16w+16). Both GEMMs use V_WMMA_F32_16X16X4_F32 (exact fp32).
//
// out[m][n] = X[m][:] . W[n][:]  ->  A = X tile (16xK), B[k][n] = W[n][k]
// A frag (ISA 7.12.2, 32-bit A 16x4): lane -> M = lane%16, K = 2*(lane/16)+{0,1}
// C/D frag: VGPR r -> M = r + 8*(lane/16), N = lane%16
// ---------------------------------------------------------------------------
__global__ void __launch_bounds__(256)
fused_gemm_norm_kernel(const float* __restrict__ node_fts,
                       const float* __restrict__ W_self,
                       const float* __restrict__ W_neigh,
                       const float* __restrict__ acc,
                       const float* __restrict__ cnt,
                       float* __restrict__ out) {
    __shared__ float tile[16 * DIM];   // 8 KB staging of the 16x128 output rows
    __shared__ float norm2[16];

    const int tid = threadIdx.x;
    if (tid < 16) norm2[tid] = 0.0f;   // zeroed before the barrier below

    const int wave  = tid >> 5;
    const int lane  = tid & 31;
    const int lm    = lane & 15;       // row within tile (A) / col within tile (B)
    const int half  = lane >> 4;       // K sub-phase for A/B fragments
    const int node0 = blockIdx.x * 16;
    const int ncol  = wave * 16 + lm;  // output column handled by this lane's B

    // per-row 1/max(count,1): row M = lm for both lane halves
    const float invc = 1.0f / fmaxf(cnt[node0 + lm], 1.0f);

    const float* xs = node_fts + (size_t)(node0 + lm) * DIM;  // self A rows
    const float* xa = acc      + (size_t)(node0 + lm) * DIM;  // neigh A rows (pre-mean)
    const float* ws = W_self   + (size_t)ncol * DIM;          // B cols = W rows
    const float* wn = W_neigh  + (size_t)ncol * DIM;

    v8f acc_s = {};   // self GEMM accumulator  (16x16 tile)
    v8f acc_n = {};   // neigh GEMM accumulator (16x16 tile)

    #pragma unroll 4
    for (int k = 0; k < DIM; k += 4) {
        const int kk = k + 2 * half;
        v2f a_s = { xs[kk],        xs[kk + 1] };
        v2f a_n = { xa[kk] * invc, xa[kk + 1] * invc };  // fold mean into A
        v2f b_s = { ws[kk],        ws[kk + 1] };
        v2f b_n = { wn[kk],        wn[kk + 1] };
        acc_s = __builtin_amdgcn_wmma_f32_16x16x4_f32(
                    false, a_s, false, b_s, (short)0, acc_s, false, false);
        acc_n = __builtin_amdgcn_wmma_f32_16x16x4_f32(
                    false, a_n, false, b_n, (short)0, acc_n, false, false);
    }

    // epilogue: neigh = LeakyReLU(neigh); o = LeakyReLU(self + neigh) -> LDS
    #pragma unroll
    for (int r = 0; r < 8; ++r) {
        float nv = acc_n[r];
        nv = (nv > 0.0f) ? nv : NEG_SLOPE * nv;
        float o = acc_s[r] + nv;
        o = (o > 0.0f) ? o : NEG_SLOPE * o;
        const int row = r + 8 * half;
        tile[row * DIM + ncol] = o;
    }
    __syncthreads();

    // row L2 norms: thread t -> row t/16, 8-col chunk (t%16)*8
    const int r  = tid >> 4;
    const int c0 = (tid & 15) * 8;
    float p = 0.0f;
    #pragma unroll
    for (int j = 0; j < 8; ++j) {
        const float v = tile[r * DIM + c0 + j];
        p += v * v;
    }
    atomicAdd(&norm2[r], p);
    __syncthreads();

    const float inv = 1.0f / fmaxf(sqrtf(norm2[r]), L2EPS);
    float* orow = out + (size_t)(node0 + r) * DIM + c0;
    #pragma unroll
    for (int j = 0; j < 8; ++j) orow[j] = tile[r * DIM + c0 + j] * inv;
}

// ---------------------------------------------------------------------------
extern "C" void kernel_launch(void* const* d_in, const int* in_sizes, int n_in,
                              void* d_out, int out_size, void* d_ws, size_t ws_size,
                              hipStream_t stream) {
    const float*     node_fts = (const float*)d_in[0];
    // d_in[1] = edge_fts (unused in 'mean' mode)
    const long long* edges    = (const long long*)d_in[2];
    const float*     W_self   = (const float*)d_in[3];
    const float*     W_neigh  = (const float*)d_in[4];
    float*           out      = (float*)d_out;

    float* acc = (float*)d_ws;                 // [N_NODES * DIM]
    float* cnt = acc + (size_t)N_NODES * DIM;  // [N_NODES]

    // 1) zero accumulator + counts
    const int total = N_NODES * DIM + N_NODES;
    zero_ws_kernel<<<(total + 255) / 256, 256, 0, stream>>>(acc, total);

    // 2) scatter-add: one wave per directed edge
    const long long nthreads = (long long)2 * N_EDGES * 32;
    const int nblocks = (int)((nthreads + 255) / 256);
    scatter_accum_kernel<<<nblocks, 256, 0, stream>>>(node_fts, edges, acc, cnt);

    // 3) fused WMMA GEMMs + activation + L2 normalize (50000 = 3125 * 16)
    fused_gemm_norm_kernel<<<N_NODES / 16, 256, 0, stream>>>(
        node_fts, W_self, W_neigh, acc, cnt, out);
}